// SCG_block_44074954391571
// MI455X (gfx1250) — compile-verified
//
#include <hip/hip_runtime.h>
#include <hip/hip_bf16.h>

typedef __attribute__((ext_vector_type(2))) float v2f;
typedef __attribute__((ext_vector_type(8))) float v8f;

#define NB 4
#define NC 256
#define NP 512      // nodes (8^3)
#define NH 9        // hidden channels

// ---- d_out layout (floats): (A, gx, loss, z_hat) concatenated flat ----
#define A_SZ     (NB*NP*NP)          // 1048576
#define GX_OFF   (A_SZ)
#define GX_SZ    (NB*NC*NP)          // 524288
#define LOSS_OFF (GX_OFF+GX_SZ)      // 1572864
#define ZH_OFF   (LOSS_OFF+1)        // 1572865, size 18432

// ---- workspace layout (floats) ----
#define WS_MU    0                   // NB*NH*NP = 18432   (b, ch, pos) natural conv layout
#define WS_LV    18432               // NB*NH*NP
#define WS_ZT    36864               // NB*16*NP = 32768   zT[b][k][n] = z[b][n][k], zero-padded k=9..15
#define WS_AD    69632               // NB*NP
#define WS_KL    71680               // NB*NP
#define WS_GAMA  73728               // NB
#define WS_GM    73732               // 1
#define WS_D     73736               // NB*NP (d^-1/2 vector)

// ------------------------------------------------------------------
// 1) AdaptiveAvgPool3d (32^3 -> 8^3): the only HBM-bandwidth pass.
//    One thread per gx element; 16x b128 loads (4x4 rows of float4).
// ------------------------------------------------------------------
__global__ void k_pool(const float* __restrict__ x, float* __restrict__ gx) {
    int t  = blockIdx.x * 256 + threadIdx.x;          // < 524288
    int ow = t & 7, oh = (t >> 3) & 7, od = (t >> 6) & 7;
    int bc = t >> 9;                                   // b*256 + c
    const float4* xp = (const float4*)(x + (size_t)bc * 32768
                                         + od * 4096 + oh * 128 + ow * 4);
    float s = 0.f;
    #pragma unroll
    for (int dd = 0; dd < 4; ++dd)
        #pragma unroll
        for (int hh = 0; hh < 4; ++hh) {
            float4 v = xp[dd * 256 + hh * 8];          // float4 units: d-stride 1024f, h-stride 32f
            s += v.x + v.y + v.z + v.w;
        }
    gx[t] = s * (1.0f / 64.0f);
}

// ------------------------------------------------------------------
// 2) conv3d 3x3x3 pad=1 (mu) and 1x1x1 (log_var). K=6912 per output.
//    (b, o) are wave-uniform -> weight loads become scalar loads;
//    gx reads coalesced over n. 0.26 GFLOP total, L2-resident.
// ------------------------------------------------------------------
__global__ void k_conv(const float* __restrict__ gx, const float* __restrict__ w_mu,
                       const float* __restrict__ b_mu, const float* __restrict__ w_lv,
                       const float* __restrict__ b_lv, float* __restrict__ mu,
                       float* __restrict__ lv) {
    int t = blockIdx.x * 256 + threadIdx.x;            // < 18432
    int n = t & 511;
    int r = t >> 9;
    int o = r % NH;
    int b = r / NH;
    int w = n & 7, h = (n >> 3) & 7, d = n >> 6;
    const float* gxb = gx + (size_t)b * NC * NP;
    const float* wmo = w_mu + o * NC * 27;
    const float* wlo = w_lv + o * NC;
    float acc  = b_mu[o];
    float accl = b_lv[o];
    for (int c = 0; c < NC; ++c) {
        const float* gxc = gxb + c * NP;
        const float* wc  = wmo + c * 27;
        accl += wlo[c] * gxc[n];
        #pragma unroll
        for (int kd = 0; kd < 3; ++kd) {
            int zd = d + kd - 1;
            if ((unsigned)zd > 7u) continue;
            #pragma unroll
            for (int kh = 0; kh < 3; ++kh) {
                int zh = h + kh - 1;
                if ((unsigned)zh > 7u) continue;
                #pragma unroll
                for (int kw = 0; kw < 3; ++kw) {
                    int zw = w + kw - 1;
                    if ((unsigned)zw > 7u) continue;
                    acc += wc[kd * 9 + kh * 3 + kw] * gxc[zd * 64 + zh * 8 + zw];
                }
            }
        }
    }
    mu[b * 4608 + o * 512 + n] = acc;
    lv[b * 4608 + o * 512 + n] = accl;
}

// ------------------------------------------------------------------
// 3) Per-(b,n): Ad = ||z_n||^2, kl partial, and build zT (padded,
//    transposed z for coalesced WMMA fragment loads).
//    NOTE: z = mu.reshape(B,512,9) is a RAW reshape: z[n][k]=mu_flat[n*9+k].
// ------------------------------------------------------------------
__global__ void k_stats(const float* __restrict__ mu, const float* __restrict__ lv,
                        float* __restrict__ zT, float* __restrict__ Ad,
                        float* __restrict__ klpn) {
    int t = blockIdx.x * 256 + threadIdx.x;            // < 2048
    int n = t & 511, b = t >> 9;
    const float* mub = mu + b * 4608;
    const float* lvb = lv + b * 4608;
    float* zTb = zT + b * 16 * 512;
    float ad = 0.f, kl = 0.f;
    #pragma unroll
    for (int k = 0; k < 16; ++k) {
        float zv = 0.f;
        if (k < NH) {
            float m = mub[n * 9 + k];
            float l = lvb[n * 9 + k];
            zv = m;
            ad += m * m;
            kl += 1.0f + 2.0f * l - m * m - expf(2.0f * l);
        }
        zTb[k * 512 + n] = zv;
    }
    Ad[t]   = ad;
    klpn[t] = kl;
}

// ------------------------------------------------------------------
// 4) Deterministic scalar reductions: gama[b], gama_mean, loss.
// ------------------------------------------------------------------
__global__ void k_scalars(const float* __restrict__ Ad, const float* __restrict__ klpn,
                          float* __restrict__ gama, float* __restrict__ gmean,
                          float* __restrict__ loss_out) {
    __shared__ float sAd[512], sDl[512], sKl[512];
    int t = threadIdx.x;                               // 512 threads
    int b = t >> 7;
    int base = b * 512 + (t & 127) * 4;
    float a = 0.f, dl = 0.f, kl = 0.f;
    #pragma unroll
    for (int k = 0; k < 4; ++k) {
        float ad = Ad[base + k];
        a += ad;
        if (ad < 1.0f) dl += logf(ad + 1e-7f);
        kl += klpn[base + k];
    }
    sAd[t] = a; sDl[t] = dl; sKl[t] = kl;
    __syncthreads();
    if (t == 0) {
        float gsum = 0.f;
        for (int bb = 0; bb < NB; ++bb) {
            float s = 0.f;
            for (int i = 0; i < 128; ++i) s += sAd[bb * 128 + i];
            float mean_d = s * (1.0f / 512.0f);
            float g = sqrtf(1.0f + 1.0f / mean_d);
            gama[bb] = g;
            gsum += g;
        }
        float gm = gsum * 0.25f;
        gmean[0] = gm;
        float dls = 0.f, kls = 0.f;
        for (int i = 0; i < 512; ++i) { dls += sDl[i]; kls += sKl[i]; }
        float dl_loss = gm * dls / (4.0f * 512.0f * 512.0f);
        float kl_loss = (-0.5f / 512.0f) * (kls / (4.0f * 512.0f));
        loss_out[0] = kl_loss - dl_loss;
    }
}

// ------------------------------------------------------------------
// 5) z_hat = gama_mean * mu * (1 - log_var); same raw reshape for mu
//    and log_var -> pure elementwise on the flat buffers.
// ------------------------------------------------------------------
__global__ void k_zhat(const float* __restrict__ mu, const float* __restrict__ lv,
                       const float* __restrict__ gmean, float* __restrict__ zh) {
    int t = blockIdx.x * 256 + threadIdx.x;            // < 18432
    zh[t] = gmean[0] * mu[t] * (1.0f - lv[t]);
}

// ------------------------------------------------------------------
// 6) Gram matrix via fp32 WMMA: A = leaky(z z^T) + diag(gama*Ad + 1).
//    One wave per 16x16 tile; K=9 padded to 12 -> 3 x v_wmma_f32_16x16x4_f32.
//    A-frag: lane l<16 holds row i0+l K={k0,k0+1}; lanes 16-31 K={k0+2,k0+3}.
//    B-frag: same with columns j0+l (B = z^T so B[k][j] = z[j][k] = zT[k][j]).
// ------------------------------------------------------------------
__global__ void k_gram(const float* __restrict__ zT, const float* __restrict__ Ad,
                       const float* __restrict__ gama, float* __restrict__ A) {
    int wid  = (blockIdx.x * 256 + threadIdx.x) >> 5;  // 0..4095
    int lane = threadIdx.x & 31;
    int b    = wid >> 10;
    int rem  = wid & 1023;
    int i0   = (rem >> 5) << 4;
    int j0   = (rem & 31) << 4;
    const float* zb = zT + b * 16 * 512;
    int half = lane >> 4;
    int li   = lane & 15;
    v8f c = {0.f, 0.f, 0.f, 0.f, 0.f, 0.f, 0.f, 0.f};
    #pragma unroll
    for (int kc = 0; kc < 3; ++kc) {
        int k0 = kc * 4 + half * 2;
        v2f av, bv;
        av.x = zb[k0 * 512 + i0 + li];
        av.y = zb[(k0 + 1) * 512 + i0 + li];
        bv.x = zb[k0 * 512 + j0 + li];
        bv.y = zb[(k0 + 1) * 512 + j0 + li];
        c = __builtin_amdgcn_wmma_f32_16x16x4_f32(false, av, false, bv,
                                                  (short)0, c, false, false);
    }
    float gm = gama[b];
    float* Ab = A + (size_t)b * NP * NP;
    const float* Adb = Ad + b * NP;
    #pragma unroll
    for (int r = 0; r < 8; ++r) {
        int i = i0 + r + half * 8;                     // C/D layout: VGPR r -> M=r (+8 for hi half)
        int j = j0 + li;
        float v = c[r];
        v = v > 0.f ? v : 0.2f * v;                    // LeakyReLU(0.2)
        if (i == j) v += gm * Adb[i] + 1.0f;           // + gama*diag(Ad) + I
        Ab[i * 512 + j] = v;
    }
}

// ------------------------------------------------------------------
// 7) d = (rowsum(A) + 1e-5)^-1/2 ; one wave per row, shfl_xor reduce.
// ------------------------------------------------------------------
__global__ void k_rowsum(const float* __restrict__ A, float* __restrict__ dvec) {
    int wid  = (blockIdx.x * 256 + threadIdx.x) >> 5;  // row 0..2047 (A symmetric)
    int lane = threadIdx.x & 31;
    const float* row = A + (size_t)wid * 512;
    float s = 0.f;
    #pragma unroll
    for (int k = 0; k < 16; ++k) s += row[lane + k * 32];
    #pragma unroll
    for (int off = 16; off; off >>= 1) s += __shfl_xor(s, off, 32);
    if (lane == 0) dvec[wid] = 1.0f / sqrtf(s + 1e-5f);
}

// ------------------------------------------------------------------
// 8) A[i][j] *= d[i]*d[j], float4 vectorized in-place.
// ------------------------------------------------------------------
__global__ void k_scale(float* __restrict__ A, const float* __restrict__ dvec) {
    int t  = blockIdx.x * 256 + threadIdx.x;           // < 262144 (float4 units)
    int j4 = t & 127;
    int i  = (t >> 7) & 511;
    int b  = t >> 16;
    float di = dvec[b * 512 + i];
    float4 dj = ((const float4*)(dvec + b * 512))[j4];
    float4* Ap = (float4*)A;
    float4 v = Ap[t];
    v.x *= di * dj.x; v.y *= di * dj.y; v.z *= di * dj.z; v.w *= di * dj.w;
    Ap[t] = v;
}

extern "C" void kernel_launch(void* const* d_in, const int* in_sizes, int n_in,
                              void* d_out, int out_size, void* d_ws, size_t ws_size,
                              hipStream_t stream) {
    (void)in_sizes; (void)n_in; (void)out_size; (void)ws_size;
    const float* x    = (const float*)d_in[0];
    const float* w_mu = (const float*)d_in[1];
    const float* b_mu = (const float*)d_in[2];
    const float* w_lv = (const float*)d_in[3];
    const float* b_lv = (const float*)d_in[4];
    float* out = (float*)d_out;
    float* ws  = (float*)d_ws;

    float* A    = out;
    float* gx   = out + GX_OFF;
    float* loss = out + LOSS_OFF;
    float* zh   = out + ZH_OFF;

    float* mu    = ws + WS_MU;
    float* lv    = ws + WS_LV;
    float* zT    = ws + WS_ZT;
    float* Ad    = ws + WS_AD;
    float* klpn  = ws + WS_KL;
    float* gama  = ws + WS_GAMA;
    float* gmean = ws + WS_GM;
    float* dvec  = ws + WS_D;

    k_pool   <<<2048, 256, 0, stream>>>(x, gx);
    k_conv   <<<  72, 256, 0, stream>>>(gx, w_mu, b_mu, w_lv, b_lv, mu, lv);
    k_stats  <<<   8, 256, 0, stream>>>(mu, lv, zT, Ad, klpn);
    k_scalars<<<   1, 512, 0, stream>>>(Ad, klpn, gama, gmean, loss);
    k_zhat   <<<  72, 256, 0, stream>>>(mu, lv, gmean, zh);
    k_gram   <<< 512, 256, 0, stream>>>(zT, Ad, gama, A);
    k_rowsum <<< 256, 256, 0, stream>>>(A, dvec);
    k_scale  <<<1024, 256, 0, stream>>>(A, dvec);
}